// UniquenessLoss_44693429682221
// MI455X (gfx1250) — compile-verified
//
#include <hip/hip_runtime.h>
#include <hip/hip_bf16.h>

// Problem constants (from reference setup_inputs): B=16, N=4096, n_u=n_v=64.
#define BATCH   16
#define NPTS    4096
#define NLOG    64          // logits per point (n_u == n_v == 64)
#define TILE    64          // points per workgroup
#define ROWP    68          // padded LDS row stride in dwords (64 data + 4 pad)
#define NKEY    (63 * 63)   // (u_pred-1, v_pred-1) key space, both in [1,63]
#define NBINS   (BATCH * NKEY)

typedef __attribute__((ext_vector_type(4))) unsigned int v4u;
typedef __attribute__((ext_vector_type(8))) int          v8i;
typedef __attribute__((ext_vector_type(4))) int          v4i;

// ---------------------------------------------------------------------------
// TDM: DMA a [64 x 64] float tile (contiguous rows, stride 64 floats) from
// global memory into LDS at lds_byte_addr, inserting 4 pad DWORDs after every
// 64 data DWORDs so each row lands at a 68-DWORD (272 B, 16B-aligned) stride.
// D# layout per CDNA5 ISA §8.3 (group 0) / §8.4 (group 1). 2-D tensor, so
// groups 2/3 are zero.
// ---------------------------------------------------------------------------
__device__ __forceinline__ void tdm_load_tile64x64(const float* gptr,
                                                   unsigned int lds_byte_addr) {
  unsigned long long ga = (unsigned long long)(size_t)gptr;

  v4u g0;
  g0.x = 1u;                                          // count=1, is_restore=0, gather off
  g0.y = lds_byte_addr;                               // lds_addr [63:32]
  g0.z = (unsigned int)(ga & 0xFFFFFFFFu);            // global_addr [95:64]
  g0.w = (unsigned int)((ga >> 32) & 0x01FFFFFFu)     // global_addr [120:96]
       | (2u << 30);                                  // type=2 ("image")

  v8i g1;
  // data_size=2 (4B) <<16 | pad_enable<<20 | pad_interval=5 (64 dwords) <<22
  // | pad_amount=3 (4 dwords) <<25
  g1[0] = (int)((2u << 16) | (1u << 20) | (5u << 22) | (3u << 25));
  g1[1] = (int)(64u << 16);   // tensor_dim0[15:0]=64 @bits[63:48]; atomic_barrier_addr=0
  g1[2] = (int)(64u << 16);   // tensor_dim1[15:0]=64 @bits[95:80]; tensor_dim0 hi=0
  g1[3] = (int)(64u << 16);   // tile_dim0=64 @bits[127:112]; tensor_dim1 hi=0
  g1[4] = 64;                 // tile_dim1=64 @bits[143:128]; tile_dim2=0 (2-D)
  g1[5] = 64;                 // tensor_dim0_stride[31:0]=64 elements
  g1[6] = 0;                  // stride hi / tensor_dim1_stride lo (unused, 2-D)
  g1[7] = 0;

  v4i z = {0, 0, 0, 0};
#if defined(__clang_major__) && (__clang_major__ >= 23)
  v8i z8 = {0, 0, 0, 0, 0, 0, 0, 0};
  __builtin_amdgcn_tensor_load_to_lds(g0, g1, z, z, z8, 0);
#else
  __builtin_amdgcn_tensor_load_to_lds(g0, g1, z, z, 0);
#endif
}

// ---------------------------------------------------------------------------
// Kernel 1: zero the per-batch histograms in workspace.
// ---------------------------------------------------------------------------
__global__ void uniq_zero_hist(int* __restrict__ hist) {
  int i = blockIdx.x * blockDim.x + threadIdx.x;
  if (i < NBINS) hist[i] = 0;
}

// ---------------------------------------------------------------------------
// Kernel 2: per-point dual argmax + histogram.
// Grid: BATCH * (NPTS/TILE) = 1024 workgroups of 64 threads (2 wave32).
// Wave 0 DMAs the u/v tiles into LDS via TDM, waits TENSORcnt, barrier, then
// each thread argmaxes its padded LDS row and atomically bumps its key bin.
// ---------------------------------------------------------------------------
__global__ void uniq_hist_kernel(const float* __restrict__ u,
                                 const float* __restrict__ v,
                                 const int* __restrict__ mask,
                                 int* __restrict__ hist) {
  __shared__ __align__(16) float u_tile[TILE * ROWP];
  __shared__ __align__(16) float v_tile[TILE * ROWP];

  const int wg    = blockIdx.x;
  const int b     = wg / (NPTS / TILE);
  const int chunk = wg % (NPTS / TILE);
  const long long pbase = (long long)b * NPTS + (long long)chunk * TILE;

  if (threadIdx.x < 32) {  // wave 0 only: TDM is a per-wave DMA op
    tdm_load_tile64x64(u + pbase * NLOG, (unsigned int)(size_t)(void*)u_tile);
    tdm_load_tile64x64(v + pbase * NLOG, (unsigned int)(size_t)(void*)v_tile);
    __builtin_amdgcn_s_wait_tensorcnt(0);
  }
  __syncthreads();

  const int p = threadIdx.x;  // one point per thread, p in [0,64)

  // argmax over u row (first-max-index semantics: strict '>' while scanning
  // in ascending index order, matching jnp.argmax).
  const float4* urow = (const float4*)(u_tile + p * ROWP);
  float ubest = -3.402823466e+38f;
  int   ui    = 0;
#pragma unroll
  for (int j = 0; j < NLOG / 4; ++j) {
    float4 q = urow[j];
    if (q.x > ubest) { ubest = q.x; ui = 4 * j + 0; }
    if (q.y > ubest) { ubest = q.y; ui = 4 * j + 1; }
    if (q.z > ubest) { ubest = q.z; ui = 4 * j + 2; }
    if (q.w > ubest) { ubest = q.w; ui = 4 * j + 3; }
  }

  const float4* vrow = (const float4*)(v_tile + p * ROWP);
  float vbest = -3.402823466e+38f;
  int   vi    = 0;
#pragma unroll
  for (int j = 0; j < NLOG / 4; ++j) {
    float4 q = vrow[j];
    if (q.x > vbest) { vbest = q.x; vi = 4 * j + 0; }
    if (q.y > vbest) { vbest = q.y; vi = 4 * j + 1; }
    if (q.z > vbest) { vbest = q.z; vi = 4 * j + 2; }
    if (q.w > vbest) { vbest = q.w; vi = 4 * j + 3; }
  }

  const int m = mask[pbase + p];
  if (m != 0 && ui > 0 && vi > 0) {
    const int key = (ui - 1) * 63 + (vi - 1);
    atomicAdd(&hist[b * NKEY + key], 1);
  }
}

// ---------------------------------------------------------------------------
// Kernel 3: loss = WEIGHT * (sum_k c_k*(c_k-1) / 2) / BATCH, exact in u64.
// ---------------------------------------------------------------------------
__global__ void uniq_finalize(const int* __restrict__ hist,
                              float* __restrict__ out) {
  __shared__ unsigned long long sdata[256];
  const int tid = threadIdx.x;

  unsigned long long acc = 0;
  for (int i = tid; i < NBINS; i += 256) {
    unsigned long long c = (unsigned long long)(unsigned int)hist[i];
    acc += c * (c - 1ull);  // c==0 -> 0 * huge == 0, exact
  }
  sdata[tid] = acc;
  __syncthreads();
#pragma unroll
  for (int s = 128; s > 0; s >>= 1) {
    if (tid < s) sdata[tid] += sdata[tid + s];
    __syncthreads();
  }
  if (tid == 0) {
    double total = (double)sdata[0];
    out[0] = (float)(0.1 * (total * 0.5) / (double)BATCH);
  }
}

// ---------------------------------------------------------------------------
extern "C" void kernel_launch(void* const* d_in, const int* in_sizes, int n_in,
                              void* d_out, int out_size, void* d_ws, size_t ws_size,
                              hipStream_t stream) {
  const float* u    = (const float*)d_in[0];
  const float* v    = (const float*)d_in[1];
  const int*   mask = (const int*)d_in[2];
  int*   hist = (int*)d_ws;            // BATCH*NKEY ints = 254 KB scratch
  float* out  = (float*)d_out;

  uniq_zero_hist<<<(NBINS + 255) / 256, 256, 0, stream>>>(hist);
  uniq_hist_kernel<<<BATCH * (NPTS / TILE), TILE, 0, stream>>>(u, v, mask, hist);
  uniq_finalize<<<1, 256, 0, stream>>>(hist, out);
}